// PDELoss_75187697483993
// MI455X (gfx1250) — compile-verified
//
#include <hip/hip_runtime.h>
#include <hip/hip_bf16.h>
#include <stdint.h>

// ---- CDNA5 TDM descriptor group vector types (amdgpu-toolchain / clang-23 6-arg form)
typedef unsigned int u32x4 __attribute__((ext_vector_type(4)));
typedef int          i32x8 __attribute__((ext_vector_type(8)));
typedef int          i32x4 __attribute__((ext_vector_type(4)));

#define BB   256
#define HH   256
#define WW   256
#define OH   254
#define OW   254
#define TH   16                        // blur output rows per band
#define NBANDS 16                      // 254/16 -> 16 bands (last has 14 rows)
#define NBPB 4                         // bands per block (pipelined)
#define NBG  (NBANDS / NBPB)           // 4 band-groups -> grid.x

// Issue one 2D TDM tile load: nrows x 256 f32 rows (stride 256) -> LDS at ldsa.
__device__ __forceinline__ void tdm_load_rows(const float* gbase, uint32_t ldsa, int nrows)
{
    const uint64_t gaddr = (uint64_t)(uintptr_t)gbase;

    u32x4 g0;
    g0[0] = 1u;                                   // count=1 (valid descriptor), user mode
    g0[1] = ldsa;                                 // lds_addr (bits 63:32)
    g0[2] = (uint32_t)gaddr;                      // global_addr[31:0]
    g0[3] = (uint32_t)((gaddr >> 32) & 0x01FFFFFFu) | (2u << 30); // addr[56:32] | type=2

    i32x8 g1;
    g1[0] = (2 << 16);       // data_size = 2 (4 bytes); mask=0; no pad/iterate/barrier
    g1[1] = (WW << 16);      // tensor_dim0[15:0] = 256   (bits 63:48)
    g1[2] = (HH << 16);      // tensor_dim1[15:0] = 256   (bits 95:80); dim0 hi = 0
    g1[3] = (WW << 16);      // tile_dim0 = 256           (bits 127:112); dim1 hi = 0
    g1[4] = nrows;           // tile_dim1 = rows          (bits 143:128); tile_dim2 = 0
    g1[5] = WW;              // tensor_dim0_stride = 256  (bits 191:160)
    g1[6] = 0;
    g1[7] = 0;

    i32x4 gz4 = {0, 0, 0, 0};
    i32x8 gz8 = {0, 0, 0, 0, 0, 0, 0, 0};
    __builtin_amdgcn_tensor_load_to_lds(g0, g1, gz4, gz4, gz8, 0);
}

__global__ __launch_bounds__(256)
void gs_fused_kernel(const float* __restrict__ pred,
                     const float* __restrict__ rhs,
                     const float* __restrict__ Lk_all,
                     const float* __restrict__ Dk_all,
                     const float* __restrict__ RR,
                     const float* __restrict__ ZZ,
                     float* __restrict__ partials)
{
    __shared__ float sPred[2][TH + 4][WW];  // double-buffered pred tiles (2 x 20 rows)
    __shared__ float sGS[TH + 2][256];      // GS rows, +1 col shift; cols 0,255 = 0
    __shared__ float sRed[256];

    const int bg  = blockIdx.x;             // band group: bands bg*NBPB .. bg*NBPB+3
    const int b   = blockIdx.y;             // batch
    const int tid = threadIdx.x;

    const size_t bOff = (size_t)b * HH * WW;
    const float* predB = pred + bOff;

    // ---- Per-batch scalars & filter taps (uniform broadcast loads)
    float Lk[9], Dk[9];
#pragma unroll
    for (int i = 0; i < 9; ++i) { Lk[i] = Lk_all[b * 9 + i]; Dk[i] = Dk_all[b * 9 + i]; }
    const float hr  = RR[bOff + 1 * WW + 2] - RR[bOff + 1 * WW + 1];
    const float hz  = ZZ[bOff + 2 * WW + 1] - ZZ[bOff + 1 * WW + 1];
    const float hr2 = hr * hr, hz2 = hz * hz;
    const float scale = (-2.0f * (hr2 + hz2)) / (hr2 * hz2);

    // Band geometry helpers
    auto bandR0   = [&](int j) { return (bg * NBPB + j) * TH; };
    auto bandRows = [&](int j) { return min(TH, OH - bandR0(j)); };
    auto bandGsLo = [&](int j) { return max(bandR0(j) - 1, 0); };
    auto bandNPred= [&](int j) {
        const int gsHi = min(bandR0(j) + bandRows(j), OH - 1);
        return (gsHi - bandGsLo(j) + 1) + 2;            // nGS + 2 <= TH+4
    };

    // ---- Pipeline prologue: TDM load for band 0 of this group
    if (tid < 32) {
        tdm_load_rows(predB + (size_t)bandGsLo(0) * WW,
                      (uint32_t)(uintptr_t)&sPred[0][0][0], bandNPred(0));
    }

    float acc = 0.0f;

    for (int j = 0; j < NBPB; ++j) {
        const int buf    = j & 1;
        const int r0     = bandR0(j);
        const int rows   = bandRows(j);
        const int gsLo   = bandGsLo(j);
        const int gsHi   = min(r0 + rows, OH - 1);
        const int nGS    = gsHi - gsLo + 1;
        const int predLo = gsLo;

        // Issue next band's DMA, then wait for the current one (in-order completion).
        if (tid < 32) {
            if (j + 1 < NBPB) {
                tdm_load_rows(predB + (size_t)bandGsLo(j + 1) * WW,
                              (uint32_t)(uintptr_t)&sPred[buf ^ 1][0][0], bandNPred(j + 1));
                __builtin_amdgcn_s_wait_tensorcnt((short)1);   // current tile landed
            } else {
                __builtin_amdgcn_s_wait_tensorcnt((short)0);   // last tile landed
            }
        }
        __syncthreads();   // tile in sPred[buf] visible; sGS free from previous iter

        // ---- Phase 1: GS_ope rows (with halo) into LDS, +1 column shift
        for (int gi = 0; gi < nGS; ++gi) {
            const int gy = gsLo + gi;
            const int pr = gy - predLo;
            if (tid < OW) {
                const int gx = tid;
                float L = 0.0f, D = 0.0f;
#pragma unroll
                for (int ky = 0; ky < 3; ++ky)
#pragma unroll
                    for (int kx = 0; kx < 3; ++kx) {
                        const float p = sPred[buf][pr + ky][gx + kx];
                        L = fmaf(Lk[ky * 3 + kx], p, L);
                        D = fmaf(Dk[ky * 3 + kx], p, D);
                    }
                // GS = Lpsi + convD/RR_in, then * scale
                const float rrv = RR[bOff + (size_t)(gy + 1) * WW + (gx + 1)];
                sGS[gi][gx + 1] = (L + D / rrv) * scale;
            }
            if (tid == 0)   sGS[gi][0]   = 0.0f;   // SAME-pad left
            if (tid == 255) sGS[gi][255] = 0.0f;   // SAME-pad right
        }
        __syncthreads();   // pred tile free for reuse (band j+2); sGS complete

        // ---- Phase 2: 3x3 Gaussian blur + squared error accumulation
        if (tid < OW) {
            const int gx = tid;
            for (int ri = 0; ri < rows; ++ri) {
                const int ry = r0 + ri;
                float s = 0.0f;
#pragma unroll
                for (int dy = -1; dy <= 1; ++dy) {
                    const int grow = ry + dy;
                    if (grow >= 0 && grow <= OH - 1) {
                        const float* rw = &sGS[grow - gsLo][gx]; // dx = -1,0,+1
                        const float w = (dy == 0) ? 2.0f : 1.0f;
                        s += w * (rw[0] + 2.0f * rw[1] + rw[2]);
                    }
                }
                s *= (1.0f / 16.0f);
                const float d = s - rhs[((size_t)b * OH + ry) * OW + gx];
                acc = fmaf(d, d, acc);
            }
        }
        __syncthreads();   // sGS free for next band
    }

    // ---- Block reduction (deterministic), one partial per block
    sRed[tid] = acc;
    __syncthreads();
#pragma unroll
    for (int off = 128; off > 0; off >>= 1) {
        if (tid < off) sRed[tid] += sRed[tid + off];
        __syncthreads();
    }
    if (tid == 0) partials[b * NBG + bg] = sRed[0];
}

__global__ __launch_bounds__(256)
void reduce_kernel(const float* __restrict__ partials, int n, float* __restrict__ out)
{
    __shared__ float sRed[256];
    float s = 0.0f;
    for (int i = threadIdx.x; i < n; i += 256) s += partials[i];  // fixed order
    sRed[threadIdx.x] = s;
    __syncthreads();
#pragma unroll
    for (int off = 128; off > 0; off >>= 1) {
        if (threadIdx.x < off) sRed[threadIdx.x] += sRed[threadIdx.x + off];
        __syncthreads();
    }
    if (threadIdx.x == 0)
        out[0] = sRed[0] / (float)((size_t)BB * OH * OW);   // mean
}

extern "C" void kernel_launch(void* const* d_in, const int* in_sizes, int n_in,
                              void* d_out, int out_size, void* d_ws, size_t ws_size,
                              hipStream_t stream) {
    const float* pred = (const float*)d_in[0];
    const float* rhs  = (const float*)d_in[1];
    const float* Lk   = (const float*)d_in[2];
    const float* Dk   = (const float*)d_in[3];
    const float* RR   = (const float*)d_in[4];
    const float* ZZ   = (const float*)d_in[5];
    float* partials   = (float*)d_ws;          // NBG*BB = 1024 floats = 4 KB scratch
    float* out        = (float*)d_out;

    dim3 grid(NBG, BB);                        // 4 band-groups x 256 batches = 1024 blocks
    gs_fused_kernel<<<grid, 256, 0, stream>>>(pred, rhs, Lk, Dk, RR, ZZ, partials);
    reduce_kernel<<<1, 256, 0, stream>>>(partials, BB * NBG, out);
}